// FocalLoss_89756226552133
// MI455X (gfx1250) — compile-verified
//
#include <hip/hip_runtime.h>
#include <hip/hip_bf16.h>
#include <stdint.h>

#ifndef __has_builtin
#define __has_builtin(x) 0
#endif

#if __has_builtin(__builtin_amdgcn_global_load_async_to_lds_b128)
#define HAVE_ASYNC_LDS 1
#else
#define HAVE_ASYNC_LDS 0
#endif

#if __has_builtin(__builtin_amdgcn_s_wait_asynccnt)
#define WAIT_ASYNC(n) __builtin_amdgcn_s_wait_asynccnt(n)
#elif HAVE_ASYNC_LDS
#define WAIT_ASYNC(n) asm volatile("s_wait_asynccnt %0" ::"i"(n) : "memory")
#else
#define WAIT_ASYNC(n) do { } while (0)
#endif

#define AS_GLOBAL __attribute__((address_space(1)))
#define AS_LDS    __attribute__((address_space(3)))

typedef int v4i_t __attribute__((ext_vector_type(4)));
typedef AS_GLOBAL v4i_t* gv4i_p;   // global (AS1) int4*
typedef AS_LDS    v4i_t* lv4i_p;   // LDS    (AS3) int4*

// ---------------------------------------------------------------------------
// Kernel 1: anchor assignment (IoU max/argmax), regression smooth-L1, num_pos
// ---------------------------------------------------------------------------
__global__ void __launch_bounds__(256)
assign_kernel(const float* __restrict__ anchors,   // (A,4)
              const float* __restrict__ regs,      // (B,A,4)
              const float* __restrict__ anns,      // (B,M,5)
              unsigned char* __restrict__ tgt,     // (B,A): 0..C-1 = pos label, 254 = neg, 255 = ignore
              double* __restrict__ reg_sum,        // (B)
              int* __restrict__ npos,              // (B)
              int A, int M)
{
    const int b   = blockIdx.y;
    const int tid = threadIdx.x;
    const int a   = blockIdx.x * blockDim.x + tid;

    __shared__ float ann[64 * 5];
    for (int i = tid; i < M * 5; i += blockDim.x)
        ann[i] = anns[(size_t)b * M * 5 + i];
    __syncthreads();

    double rsum = 0.0;
    int    pc   = 0;

    if (a < A) {
        const float ax1 = anchors[4 * a + 0], ay1 = anchors[4 * a + 1];
        const float ax2 = anchors[4 * a + 2], ay2 = anchors[4 * a + 3];
        const float aw = ax2 - ax1, ah = ay2 - ay1;
        const float areaA = aw * ah;

        float best = -1e30f;
        int   arg  = 0;
        #pragma unroll 4
        for (int m = 0; m < M; ++m) {
            const float bx1 = ann[5 * m + 0], by1 = ann[5 * m + 1];
            const float bx2 = ann[5 * m + 2], by2 = ann[5 * m + 3];
            const float lab = ann[5 * m + 4];
            float iw = fminf(ax2, bx2) - fmaxf(ax1, bx1); iw = fmaxf(iw, 0.0f);
            float ih = fminf(ay2, by2) - fmaxf(ay1, by1); ih = fmaxf(ih, 0.0f);
            const float inter = iw * ih;
            const float ua = fmaxf(areaA + (bx2 - bx1) * (by2 - by1) - inter, 1e-8f);
            float iou = inter / ua;
            if (lab == -1.0f) iou = -1.0f;
            if (iou > best) { best = iou; arg = m; }   // first-max wins (strict >)
        }

        const bool pos = best >= 0.4f;   // IOU_THR + 0.1
        const bool neg = best <  0.3f;   // IOU_THR
        unsigned char tb = pos ? (unsigned char)(int)ann[5 * arg + 4]
                               : (neg ? (unsigned char)254 : (unsigned char)255);
        tgt[(size_t)b * A + a] = tb;

        if (pos) {
            const float gx1 = ann[5 * arg + 0], gy1 = ann[5 * arg + 1];
            const float gx2 = ann[5 * arg + 2], gy2 = ann[5 * arg + 3];
            float gw = gx2 - gx1, gh = gy2 - gy1;
            const float gcx = gx1 + 0.5f * gw, gcy = gy1 + 0.5f * gh; // uses unclamped wh
            gw = fmaxf(gw, 1.0f); gh = fmaxf(gh, 1.0f);
            const float acx = ax1 + 0.5f * aw, acy = ay1 + 0.5f * ah;
            float tt[4];
            tt[0] = ((gcx - acx) / aw) / 0.1f;
            tt[1] = ((gcy - acy) / ah) / 0.1f;
            tt[2] = __logf(gw / aw) / 0.2f;
            tt[3] = __logf(gh / ah) / 0.2f;
            const float* rp = regs + ((size_t)b * A + a) * 4;
            #pragma unroll
            for (int k = 0; k < 4; ++k) {
                const float d = fabsf(tt[k] - rp[k]);
                rsum += (d <= 1.0f) ? 0.5f * d * d : d - 0.5f;
            }
            pc = 1;
        }
    }

    __shared__ double sred[256];
    __shared__ int    sint[256];
    sred[tid] = rsum;
    sint[tid] = pc;
    __syncthreads();
    for (int s = 128; s > 0; s >>= 1) {
        if (tid < s) { sred[tid] += sred[tid + s]; sint[tid] += sint[tid + s]; }
        __syncthreads();
    }
    if (tid == 0) {
        atomicAdd(&reg_sum[b], sred[0]);
        atomicAdd(&npos[b], sint[0]);
    }
}

// ---------------------------------------------------------------------------
// Kernel 2: focal classification loss over (B, A, C) — memory-bound stream.
// 4-deep async global->LDS pipeline (CDNA5 ASYNCcnt path), non-temporal
// loads (stream is read exactly once). Each thread consumes exactly the
// float4 it fetched, so only per-wave s_wait_asynccnt is needed (no barriers
// in the hot loop).
// ---------------------------------------------------------------------------
__global__ void __launch_bounds__(256)
focal_cls_kernel(const float* __restrict__ cls,
                 const unsigned char* __restrict__ tgt,
                 double* __restrict__ cls_sum,
                 int A, int C)
{
    const int b   = blockIdx.y;
    const int tid = threadIdx.x;
    const int E   = A * C;                       // 8,000,000 for this problem
    const float* __restrict__ base        = cls + (long long)b * E;
    const unsigned char* __restrict__ tg  = tgt + (size_t)b * A;

    __shared__ __align__(16) float tile[4][1024];  // 4 x 4KB circular buffer
    __shared__ double sred[256];

    const int nTiles = (E + 1023) >> 10;
    const int G      = gridDim.x;
    float acc = 0.0f;

#if HAVE_ASYNC_LDS
#define ISSUE_TILE(tt, bufidx) do {                                          \
        int _idx = ((tt) << 10) + (tid << 2);                                \
        int _c   = _idx > E - 4 ? E - 4 : _idx;                              \
        __builtin_amdgcn_global_load_async_to_lds_b128(                      \
            (gv4i_p)(base + _c), (lv4i_p)(&tile[bufidx][tid << 2]),          \
            0, 1 /* TH=NT: single-use stream */);                            \
    } while (0)
#else
#define ISSUE_TILE(tt, bufidx) do {                                          \
        int _idx = ((tt) << 10) + (tid << 2);                                \
        int _c   = _idx > E - 4 ? E - 4 : _idx;                              \
        *(float4*)(&tile[bufidx][tid << 2]) = *(const float4*)(base + _c);   \
    } while (0)
#endif

    const int t0 = blockIdx.x;
    #pragma unroll
    for (int k = 0; k < 3; ++k) {                 // prologue: 3 tiles ahead
        const int tt = t0 + k * G;
        if (tt < nTiles) ISSUE_TILE(tt, k);
    }

    int it = 0;
    for (int t = t0; t < nTiles; t += G, ++it) {
        const int cur = it & 3;
        const int nxt = (it + 3) & 3;
        if (t + 3 * G < nTiles) {                 // steady state: keep 4 in flight
            ISSUE_TILE(t + 3 * G, nxt);
            WAIT_ASYNC(3);
        } else if (t + 2 * G < nTiles) {
            WAIT_ASYNC(2);
        } else if (t + G < nTiles) {
            WAIT_ASYNC(1);
        } else {
            WAIT_ASYNC(0);
        }

        const int idx = (t << 10) + (tid << 2);
        if (idx <= E - 4) {
            const unsigned ui = (unsigned)idx;
            unsigned a, c0;
            if (C == 80) { a = ui / 80u; c0 = ui - a * 80u; }     // mul-hi path
            else         { a = ui / (unsigned)C; c0 = ui - a * (unsigned)C; }
            const unsigned char tb = tg[a];
            if (tb != 255u) {                                     // ignore-anchors: zero loss
                const float4 v = *(const float4*)(&tile[cur][tid << 2]);
                const float pv[4] = { v.x, v.y, v.z, v.w };
                #pragma unroll
                for (int k = 0; k < 4; ++k) {
                    float p = fminf(fmaxf(pv[k], 1e-4f), 1.0f - 1e-4f);
                    const bool ispos = (tb < 200u) && ((int)(c0 + k) == (int)tb);
                    const float x = ispos ? p : (1.0f - p);       // log argument
                    const float m = ispos ? (1.0f - p) : p;       // focal modulator
                    const float w = ispos ? 0.25f : 0.75f;        // alpha / (1-alpha)
                    acc -= w * m * m * __logf(x);
                }
            }
        }
    }
#undef ISSUE_TILE

    sred[tid] = (double)acc;
    __syncthreads();
    for (int s = 128; s > 0; s >>= 1) {
        if (tid < s) sred[tid] += sred[tid + s];
        __syncthreads();
    }
    if (tid == 0) atomicAdd(&cls_sum[b], sred[0]);
}

// ---------------------------------------------------------------------------
// Kernel 3: finalize scalars
// ---------------------------------------------------------------------------
__global__ void finalize_kernel(const double* __restrict__ cls_sum,
                                const double* __restrict__ reg_sum,
                                const int* __restrict__ npos,
                                float* __restrict__ out, int B)
{
    if (threadIdx.x == 0 && blockIdx.x == 0) {
        double cl = 0.0, rl = 0.0;
        long long tot = 0;
        for (int b = 0; b < B; ++b) {
            const double np = (double)npos[b];
            cl += cls_sum[b] / fmax(np, 1.0);
            rl += (npos[b] > 0) ? reg_sum[b] / fmax(np * 4.0, 1.0) : 0.0;
            tot += npos[b];
        }
        out[0] = (float)(cl / (double)B);
        out[1] = (float)(rl / (double)B);
        out[2] = (float)tot;
    }
}

// ---------------------------------------------------------------------------
// Host launcher
// ---------------------------------------------------------------------------
extern "C" void kernel_launch(void* const* d_in, const int* in_sizes, int n_in,
                              void* d_out, int out_size, void* d_ws, size_t ws_size,
                              hipStream_t stream)
{
    const float* cls     = (const float*)d_in[0];   // (B,A,C)
    const float* regs    = (const float*)d_in[1];   // (B,A,4)
    const float* anchors = (const float*)d_in[2];   // (1,A,4)
    const float* anns    = (const float*)d_in[3];   // (B,M,5)

    const int A = in_sizes[2] / 4;
    const int B = in_sizes[1] / (A * 4);
    const int C = in_sizes[0] / (B * A);
    const int M = in_sizes[3] / (B * 5);

    // workspace layout: [0,128)=cls_sum f64[16], [128,256)=reg_sum f64[16],
    //                   [256,512)=npos i32, [512,...)=targets u8[B*A]
    double*        cls_sum = (double*)d_ws;
    double*        reg_sum = (double*)((char*)d_ws + 128);
    int*           npos    = (int*)((char*)d_ws + 256);
    unsigned char* tgt     = (unsigned char*)d_ws + 512;

    (void)hipMemsetAsync(d_ws, 0, 512, stream);

    dim3 blk(256);
    dim3 g1((A + 255) / 256, B);
    assign_kernel<<<g1, blk, 0, stream>>>(anchors, regs, anns, tgt, reg_sum, npos, A, M);

    const int E      = A * C;
    const int nTiles = (E + 1023) >> 10;
    const int gx     = nTiles < 1024 ? nTiles : 1024;
    dim3 g2(gx, B);
    focal_cls_kernel<<<g2, blk, 0, stream>>>(cls, tgt, cls_sum, A, C);

    finalize_kernel<<<1, 32, 0, stream>>>(cls_sum, reg_sum, npos, (float*)d_out, B);
}